// GraphNetwork_19181323944602
// MI455X (gfx1250) — compile-verified
//
#include <hip/hip_runtime.h>
#include <hip/hip_bf16.h>
#include <math.h>

// Problem constants (match reference)
constexpr int N  = 100000;
constexpr int E  = 500000;
constexpr int F  = 128;
constexpr int H  = 64;
constexpr int R  = 16;
constexpr int NB = 30;
constexpr int G  = 1000;
constexpr int C  = 10;
constexpr int FH = F + H;   // 192

typedef __attribute__((ext_vector_type(2))) float v2f;
typedef __attribute__((ext_vector_type(8))) float v8f;

// D = A(16x4 f32) * B(4x16 f32) + C(16x16 f32)  -- full f32 precision matrix op
__device__ __forceinline__ v8f wmma_f32x4(v2f a, v2f b, v8f c) {
  return __builtin_amdgcn_wmma_f32_16x16x4_f32(
      /*neg_a=*/false, a, /*neg_b=*/false, b,
      /*c_mod=*/(short)0, c, /*reuse_a=*/false, /*reuse_b=*/false);
}

// ---------------------------------------------------------------------------
// K1: GraphConv message scatter: agg[dst] += x[src] * edge_norm
// one 32-lane group per edge, 4 floats per lane
__global__ void k_scatter_msgs(const float* __restrict__ x,
                               const int* __restrict__ src,
                               const int* __restrict__ dst,
                               const float* __restrict__ enorm,
                               float* __restrict__ agg) {
  long long t = (long long)blockIdx.x * blockDim.x + threadIdx.x;
  int e = (int)(t >> 5);
  int lane = (int)(t & 31);
  if (e >= E) return;
  int s = src[e], d = dst[e];
  float w = enorm[e];
  const float4* xr = (const float4*)(x + (size_t)s * F);
  float* ar = agg + (size_t)d * F;
  float4 v = xr[lane];
  int f0 = lane * 4;
  atomicAdd(&ar[f0 + 0], v.x * w);
  atomicAdd(&ar[f0 + 1], v.y * w);
  atomicAdd(&ar[f0 + 2], v.z * w);
  atomicAdd(&ar[f0 + 3], v.w * w);
}

// ---------------------------------------------------------------------------
// K2: h = agg @ w1_rel + x @ w1_root + b1   (WMMA f32, K=256 combined)
// block = 128 threads = 4 waves; wave w -> output cols [16w, 16w+16)
__global__ void k_gemm1(const float* __restrict__ agg, const float* __restrict__ x,
                        const float* __restrict__ w1_rel, const float* __restrict__ w1_root,
                        const float* __restrict__ b1, float* __restrict__ h) {
  int tile  = blockIdx.x;               // 16 nodes
  int wave  = threadIdx.x >> 5;
  int lane  = threadIdx.x & 31;
  int m     = lane & 15;                // A row
  int khalf = (lane >> 4) * 2;          // A/B K offset within 4
  int col   = wave * 16 + (lane & 15);  // B/C/D column
  int node  = tile * 16 + m;
  const float* arow = agg + (size_t)node * F;
  const float* xrow = x   + (size_t)node * F;
  v8f acc = {};
  #pragma unroll
  for (int k = 0; k < F; k += 4) {
    int kk = k + khalf;
    v2f a; a.x = arow[kk]; a.y = arow[kk + 1];
    v2f b; b.x = w1_rel[(size_t)kk * H + col]; b.y = w1_rel[(size_t)(kk + 1) * H + col];
    acc = wmma_f32x4(a, b, acc);
  }
  #pragma unroll
  for (int k = 0; k < F; k += 4) {
    int kk = k + khalf;
    v2f a; a.x = xrow[kk]; a.y = xrow[kk + 1];
    v2f b; b.x = w1_root[(size_t)kk * H + col]; b.y = w1_root[(size_t)(kk + 1) * H + col];
    acc = wmma_f32x4(a, b, acc);
  }
  float bias = b1[col];
  int rbase = (lane >> 4) * 8;          // C/D rows: lanes 16-31 hold M+8
  #pragma unroll
  for (int j = 0; j < 8; ++j) {
    int mm = rbase + j;
    h[(size_t)(tile * 16 + mm) * H + col] = acc[j] + bias;
  }
}

// ---------------------------------------------------------------------------
// K3: per-(dst, relation) edge counts + per-relation totals
__global__ void k_count(const int* __restrict__ dst, const int* __restrict__ etype,
                        float* __restrict__ cnt, int* __restrict__ relCount) {
  int e = blockIdx.x * blockDim.x + threadIdx.x;
  if (e >= E) return;
  int r = etype[e];
  atomicAdd(&cnt[(size_t)dst[e] * R + r], 1.0f);
  atomicAdd(&relCount[r], 1);
}

// K3b: tiny serial prefix over 16 relation counts
__global__ void k_prefix(const int* __restrict__ relCount,
                         int* __restrict__ relStart, int* __restrict__ relCursor) {
  if (threadIdx.x == 0 && blockIdx.x == 0) {
    int acc = 0;
    for (int r = 0; r < R; ++r) {
      relStart[r] = acc; relCursor[r] = acc; acc += relCount[r];
    }
    relStart[R] = acc;
  }
}

// K3c: scatter edge ids into relation bins
__global__ void k_bin(const int* __restrict__ etype, int* __restrict__ relCursor,
                      int* __restrict__ binned) {
  int e = blockIdx.x * blockDim.x + threadIdx.x;
  if (e >= E) return;
  int pos = atomicAdd(&relCursor[etype[e]], 1);
  binned[pos] = e;
}

// ---------------------------------------------------------------------------
// K4: Wrel[r] = sum_b comp[r][b] * bases[b]   (R*H*H outputs)
__global__ void k_basis(const float* __restrict__ comp, const float* __restrict__ bases,
                        float* __restrict__ Wrel) {
  int idx = blockIdx.x * blockDim.x + threadIdx.x;
  if (idx >= R * H * H) return;
  int r = idx / (H * H);
  int io = idx - r * (H * H);
  float acc = 0.f;
  #pragma unroll 6
  for (int b = 0; b < NB; ++b) acc += comp[r * NB + b] * bases[(size_t)b * H * H + io];
  Wrel[idx] = acc;
}

// ---------------------------------------------------------------------------
// K5: out = h @ w2_root + b2   (WMMA f32, K=64) -- full overwrite (no zero init)
__global__ void k_gemm_root2(const float* __restrict__ h, const float* __restrict__ w2_root,
                             const float* __restrict__ b2, float* __restrict__ out) {
  int tile = blockIdx.x;
  int wave = threadIdx.x >> 5;
  int lane = threadIdx.x & 31;
  int m = lane & 15;
  int khalf = (lane >> 4) * 2;
  int col = wave * 16 + (lane & 15);
  const float* hrow = h + (size_t)(tile * 16 + m) * H;
  v8f acc = {};
  #pragma unroll
  for (int k = 0; k < H; k += 4) {
    int kk = k + khalf;
    v2f a; a.x = hrow[kk]; a.y = hrow[kk + 1];
    v2f b; b.x = w2_root[(size_t)kk * H + col]; b.y = w2_root[(size_t)(kk + 1) * H + col];
    acc = wmma_f32x4(a, b, acc);
  }
  float bias = b2[col];
  int rbase = (lane >> 4) * 8;
  #pragma unroll
  for (int j = 0; j < 8; ++j) {
    int mm = rbase + j;
    out[(size_t)(tile * 16 + mm) * H + col] = acc[j] + bias;
  }
}

// ---------------------------------------------------------------------------
// K6: RGCN edge pass. For each relation's 16-edge tile:
//     D(16x64) = h[src(tile)] @ Wrel[r]; out[dst] += D_row / max(cnt[dst][r],1)
__global__ void k_rgcn_edges(const float* __restrict__ h,
                             const int* __restrict__ src, const int* __restrict__ dst,
                             const float* __restrict__ Wrel, const float* __restrict__ cnt,
                             const int* __restrict__ binned, const int* __restrict__ relStart,
                             float* __restrict__ out) {
  int r = blockIdx.y;
  int segStart = relStart[r], segEnd = relStart[r + 1];
  int tileBase = segStart + blockIdx.x * 16;
  if (tileBase >= segEnd) return;  // uniform early-exit
  int wave = threadIdx.x >> 5;
  int lane = threadIdx.x & 31;
  int m = lane & 15;
  int khalf = (lane >> 4) * 2;
  int col = wave * 16 + (lane & 15);
  bool valid = (tileBase + m) < segEnd;
  int e = binned[valid ? (tileBase + m) : tileBase];
  int s = src[e];
  float msk = valid ? 1.0f : 0.0f;
  const float* hrow = h + (size_t)s * H;
  const float* W = Wrel + (size_t)r * H * H;
  v8f acc = {};
  #pragma unroll
  for (int k = 0; k < H; k += 4) {
    int kk = k + khalf;
    v2f a; a.x = hrow[kk] * msk; a.y = hrow[kk + 1] * msk;
    v2f b; b.x = W[(size_t)kk * H + col]; b.y = W[(size_t)(kk + 1) * H + col];
    acc = wmma_f32x4(a, b, acc);
  }
  int rbase = (lane >> 4) * 8;
  #pragma unroll
  for (int j = 0; j < 8; ++j) {
    int idx = tileBase + rbase + j;
    if (idx < segEnd) {
      int ee = binned[idx];
      int d = dst[ee];
      float scale = 1.0f / fmaxf(cnt[(size_t)d * R + r], 1.0f);
      atomicAdd(&out[(size_t)d * H + col], acc[j] * scale);
    }
  }
}

// ---------------------------------------------------------------------------
// K7: bounds = cumsum(seq_lengths) (tiny serial)
__global__ void k_bounds(const int* __restrict__ seq, int* __restrict__ bounds) {
  if (threadIdx.x == 0 && blockIdx.x == 0) {
    int acc = 0;
    for (int g = 0; g < G; ++g) { acc += seq[g]; bounds[g] = acc; }
  }
}

// K7b: per-graph sum & max pooling over feats = [x | out]; block=192 threads
__global__ void k_pool(const float* __restrict__ x, const float* __restrict__ out,
                       const int* __restrict__ bounds,
                       float* __restrict__ gsum, float* __restrict__ gmax) {
  int g = blockIdx.x;
  int f = threadIdx.x;              // 0..191
  int start = (g == 0) ? 0 : bounds[g - 1];
  int end = bounds[g];
  if (start > N) start = N;
  if (end > N) end = N;
  float s = 0.f, mx = -INFINITY;
  for (int node = start; node < end; ++node) {
    float v = (f < F) ? x[(size_t)node * F + f]
                      : out[(size_t)node * H + (f - F)];
    s += v;
    mx = fmaxf(mx, v);
  }
  gsum[(size_t)g * FH + f] = s;
  gmax[(size_t)g * FH + f] = mx;
}

// ---------------------------------------------------------------------------
// K8: hidden = relu([gsum|gmax] @ lin_w + lin_b)  (WMMA f32, K=384)
__global__ void k_lin(const float* __restrict__ gsum, const float* __restrict__ gmax,
                      const float* __restrict__ lin_w, const float* __restrict__ lin_b,
                      float* __restrict__ hidden) {
  int tile = blockIdx.x;
  int wave = threadIdx.x >> 5;
  int lane = threadIdx.x & 31;
  int m = lane & 15;
  int khalf = (lane >> 4) * 2;
  int col = wave * 16 + (lane & 15);
  int g = tile * 16 + m;
  bool valid = g < G;
  int gc = valid ? g : 0;
  float msk = valid ? 1.0f : 0.0f;
  const float* sr = gsum + (size_t)gc * FH;
  const float* mr = gmax + (size_t)gc * FH;
  v8f acc = {};
  #pragma unroll 4
  for (int k = 0; k < 2 * FH; k += 4) {
    int kk = k + khalf;
    v2f a;
    a.x = ((kk     < FH) ? sr[kk]          : mr[kk - FH])     * msk;
    a.y = ((kk + 1 < FH) ? sr[kk + 1]      : mr[kk + 1 - FH]) * msk;
    v2f b;
    b.x = lin_w[(size_t)kk * H + col];
    b.y = lin_w[(size_t)(kk + 1) * H + col];
    acc = wmma_f32x4(a, b, acc);
  }
  float bias = lin_b[col];
  int rbase = (lane >> 4) * 8;
  #pragma unroll
  for (int j = 0; j < 8; ++j) {
    int gg = tile * 16 + rbase + j;
    if (gg < G) hidden[(size_t)gg * H + col] = fmaxf(acc[j] + bias, 0.0f);
  }
}

// ---------------------------------------------------------------------------
// K9: logits = hidden @ fc_w + fc_b; log_softmax unless *avec
__global__ void k_head(const float* __restrict__ hidden, const float* __restrict__ fc_w,
                       const float* __restrict__ fc_b, const int* __restrict__ avec,
                       float* __restrict__ outp) {
  int g = blockIdx.x * blockDim.x + threadIdx.x;
  if (g >= G) return;
  const float* hr = hidden + (size_t)g * H;
  float logit[C];
  float mx = -INFINITY;
  #pragma unroll
  for (int c = 0; c < C; ++c) {
    float a = fc_b[c];
    for (int k = 0; k < H; ++k) a += hr[k] * fc_w[(size_t)k * C + c];
    logit[c] = a;
    mx = fmaxf(mx, a);
  }
  if (*avec) {
    for (int c = 0; c < C; ++c) outp[(size_t)g * C + c] = logit[c];
    return;
  }
  float sum = 0.f;
  #pragma unroll
  for (int c = 0; c < C; ++c) sum += expf(logit[c] - mx);
  float lse = mx + logf(sum);
  #pragma unroll
  for (int c = 0; c < C; ++c) outp[(size_t)g * C + c] = logit[c] - lse;
}

// ---------------------------------------------------------------------------
extern "C" void kernel_launch(void* const* d_in, const int* in_sizes, int n_in,
                              void* d_out, int out_size, void* d_ws, size_t ws_size,
                              hipStream_t stream) {
  const float* x        = (const float*)d_in[0];
  const int*   eidx     = (const int*)  d_in[1];
  const float* enorm    = (const float*)d_in[2];
  const int*   etype    = (const int*)  d_in[3];
  const int*   seq      = (const int*)  d_in[4];
  const int*   avec     = (const int*)  d_in[5];
  const float* w1_rel   = (const float*)d_in[6];
  const float* w1_root  = (const float*)d_in[7];
  const float* b1       = (const float*)d_in[8];
  const float* bases    = (const float*)d_in[9];
  const float* comp     = (const float*)d_in[10];
  const float* w2_root  = (const float*)d_in[11];
  const float* b2       = (const float*)d_in[12];
  const float* lin_w    = (const float*)d_in[13];
  const float* lin_b    = (const float*)d_in[14];
  const float* fc_w     = (const float*)d_in[15];
  const float* fc_b     = (const float*)d_in[16];
  float* outp = (float*)d_out;

  const int* src = eidx;          // edge_index[0]
  const int* dst = eidx + E;      // edge_index[1]

  // ---- workspace carve-out (256B aligned); zeroed region comes first ----
  char* base = (char*)d_ws;
  size_t off = 0;
  auto carve = [&](size_t bytes) -> char* {
    char* p = base + off;
    off = (off + bytes + 255) & ~(size_t)255;
    return p;
  };
  float* agg      = (float*)carve((size_t)N * F * 4);   // zeroed
  float* cnt      = (float*)carve((size_t)N * R * 4);   // zeroed
  int*   relCount = (int*)  carve(R * 4);               // zeroed
  size_t zeroBytes = off;
  int*   relCursor = (int*)carve(R * 4);
  int*   relStart  = (int*)carve((R + 1) * 4);
  int*   binned    = (int*)carve((size_t)E * 4);
  float* h         = (float*)carve((size_t)N * H * 4);
  float* out_nodes = (float*)carve((size_t)N * H * 4);
  float* Wrel      = (float*)carve((size_t)R * H * H * 4);
  int*   bounds    = (int*)  carve(G * 4);
  float* gsum      = (float*)carve((size_t)G * FH * 4);
  float* gmax      = (float*)carve((size_t)G * FH * 4);
  float* hidden    = (float*)carve((size_t)G * H * 4);
  (void)ws_size; (void)n_in; (void)in_sizes; (void)out_size;

  hipMemsetAsync(d_ws, 0, zeroBytes, stream);

  // Stage 1: GraphConv
  {
    long long threads = (long long)E * 32;
    int blocks = (int)((threads + 255) / 256);
    k_scatter_msgs<<<blocks, 256, 0, stream>>>(x, src, dst, enorm, agg);
  }
  k_gemm1<<<N / 16, 128, 0, stream>>>(agg, x, w1_rel, w1_root, b1, h);

  // Stage 2: RGCN prep (counts, bins, basis weights)
  k_count<<<(E + 255) / 256, 256, 0, stream>>>(dst, etype, cnt, relCount);
  k_prefix<<<1, 32, 0, stream>>>(relCount, relStart, relCursor);
  k_bin<<<(E + 255) / 256, 256, 0, stream>>>(etype, relCursor, binned);
  k_basis<<<(R * H * H + 255) / 256, 256, 0, stream>>>(comp, bases, Wrel);

  // Stage 2: root transform then edge-tile WMMA with scaled atomic scatter
  k_gemm_root2<<<N / 16, 128, 0, stream>>>(h, w2_root, b2, out_nodes);
  {
    dim3 grid((E + 15) / 16, R);
    k_rgcn_edges<<<grid, 128, 0, stream>>>(h, src, dst, Wrel, cnt, binned,
                                           relStart, out_nodes);
  }

  // Stage 3: pooling + MLP head
  k_bounds<<<1, 32, 0, stream>>>(seq, bounds);
  k_pool<<<G, FH, 0, stream>>>(x, out_nodes, bounds, gsum, gmax);
  k_lin<<<(G + 15) / 16, 128, 0, stream>>>(gsum, gmax, lin_w, lin_b, hidden);
  k_head<<<(G + 127) / 128, 128, 0, stream>>>(hidden, fc_w, fc_b, avec, outp);
}